// ResourceEmbeddingLayer_74208444940406
// MI455X (gfx1250) — compile-verified
//
#include <hip/hip_runtime.h>
#include <math.h>

typedef __attribute__((ext_vector_type(2))) float v2f;
typedef __attribute__((ext_vector_type(8))) float v8f;

#define R_N  50000
#define O_N  100000
#define E1_N 800000
#define E2_N 400000
#define D_N  128
#define NT_N (R_N + E1_N + E2_N)
#define NPART 1024

// ---------------------------------------------------------------------------
// GEMM: Y[M x 128] = A[M x 128] @ W[128 x 128] using V_WMMA_F32_16X16X4_F32.
// Block = 256 threads = 8 waves; each block does a 16-row x 128-col tile,
// wave w owns columns [16w, 16w+16). W staged in LDS (64KB) once per block.
// A-matrix per-lane layout (16x4 f32): lane = M | (K_half<<4); VGPR0 = K(+0),
// VGPR1 = K(+1) with lanes 16-31 holding K+2/K+3. B mirrors with N across
// lanes. C/D: VGPR i holds row i (lanes 0-15) and row i+8 (lanes 16-31).
// ---------------------------------------------------------------------------

__global__ __launch_bounds__(256) void gemm_self_kernel(
    const float* __restrict__ X, const float* __restrict__ W,
    float* __restrict__ Y)
{
  __shared__ __align__(16) float Wl[128 * 128];
  for (int i = threadIdx.x * 4; i < 128 * 128; i += 1024)
    *(float4*)(&Wl[i]) = *(const float4*)(&W[i]);
  __syncthreads();

  const int t = threadIdx.x;
  const int wave = t >> 5, lane = t & 31;
  const int hi = lane >> 4, lo = lane & 15;
  const int r0 = blockIdx.x * 16;
  const int n0 = wave * 16;
  const float* arow = X + (size_t)(r0 + lo) * 128;

  v8f c = {};
  for (int k0 = 0; k0 < 128; k0 += 4) {
    const int kb = k0 + 2 * hi;
    v2f a, b;
    a.x = arow[kb];
    a.y = arow[kb + 1];
    b.x = Wl[kb * 128 + n0 + lo];
    b.y = Wl[(kb + 1) * 128 + n0 + lo];
    c = __builtin_amdgcn_wmma_f32_16x16x4_f32(false, a, false, b, (short)0, c,
                                              false, false);
  }
  float* yb = Y + (size_t)r0 * 128 + n0 + lo;
#pragma unroll
  for (int i = 0; i < 8; ++i) yb[(size_t)(i + 8 * hi) * 128] = c[i];
}

// ops_edge = concat(operations[need_src], need_edge_attr) @ W_op
// gather + concat fused into the A loads (k<96 -> operations, else eattr).
__global__ __launch_bounds__(256) void gemm_op_kernel(
    const float* __restrict__ operations, const float* __restrict__ eattr,
    const int* __restrict__ need_src, const float* __restrict__ W,
    float* __restrict__ Y)
{
  __shared__ __align__(16) float Wl[128 * 128];
  for (int i = threadIdx.x * 4; i < 128 * 128; i += 1024)
    *(float4*)(&Wl[i]) = *(const float4*)(&W[i]);
  __syncthreads();

  const int t = threadIdx.x;
  const int wave = t >> 5, lane = t & 31;
  const int hi = lane >> 4, lo = lane & 15;
  const int e0 = blockIdx.x * 16;
  const int n0 = wave * 16;
  const int e = e0 + lo;
  const float* oprow = operations + (size_t)need_src[e] * 96;
  const float* earow = eattr + (size_t)e * 32;

  v8f c = {};
  for (int k0 = 0; k0 < 128; k0 += 4) {
    const int kb = k0 + 2 * hi;
    v2f a, b;
    if (kb < 96) {        // kb is even, boundary 96 never straddled
      a.x = oprow[kb];
      a.y = oprow[kb + 1];
    } else {
      a.x = earow[kb - 96];
      a.y = earow[kb - 95];
    }
    b.x = Wl[kb * 128 + n0 + lo];
    b.y = Wl[(kb + 1) * 128 + n0 + lo];
    c = __builtin_amdgcn_wmma_f32_16x16x4_f32(false, a, false, b, (short)0, c,
                                              false, false);
  }
  float* yb = Y + (size_t)e0 * 128 + n0 + lo;
#pragma unroll
  for (int i = 0; i < 8; ++i) yb[(size_t)(i + 8 * hi) * 128] = c[i];
}

// res1 = resources[same_src] @ W_res (row gather fused into A loads)
__global__ __launch_bounds__(256) void gemm_res_kernel(
    const float* __restrict__ resources, const int* __restrict__ same_src,
    const float* __restrict__ W, float* __restrict__ Y)
{
  __shared__ __align__(16) float Wl[128 * 128];
  for (int i = threadIdx.x * 4; i < 128 * 128; i += 1024)
    *(float4*)(&Wl[i]) = *(const float4*)(&W[i]);
  __syncthreads();

  const int t = threadIdx.x;
  const int wave = t >> 5, lane = t & 31;
  const int hi = lane >> 4, lo = lane & 15;
  const int e0 = blockIdx.x * 16;
  const int n0 = wave * 16;
  const float* arow = resources + (size_t)same_src[e0 + lo] * 128;

  v8f c = {};
  for (int k0 = 0; k0 < 128; k0 += 4) {
    const int kb = k0 + 2 * hi;
    v2f a, b;
    a.x = arow[kb];
    a.y = arow[kb + 1];
    b.x = Wl[kb * 128 + n0 + lo];
    b.y = Wl[(kb + 1) * 128 + n0 + lo];
    c = __builtin_amdgcn_wmma_f32_16x16x4_f32(false, a, false, b, (short)0, c,
                                              false, false);
  }
  float* yb = Y + (size_t)e0 * 128 + n0 + lo;
#pragma unroll
  for (int i = 0; i < 8; ++i) yb[(size_t)(i + 8 * hi) * 128] = c[i];
}

// ---------------------------------------------------------------------------
// Attention logits: wave-per-row 128-wide dot + leaky_relu(0.2)
// ---------------------------------------------------------------------------
__device__ __forceinline__ float wave_sum(float p) {
#pragma unroll
  for (int m = 16; m > 0; m >>= 1) p += __shfl_xor(p, m, 32);
  return p;
}

__global__ __launch_bounds__(256) void att_self_kernel(
    const float* __restrict__ selfres, const float* __restrict__ avec,
    float* __restrict__ logits, int Rn)
{
  const int w = (blockIdx.x * blockDim.x + threadIdx.x) >> 5;
  const int lane = threadIdx.x & 31;
  if (w >= Rn) return;
  const int c = lane * 4;
  float4 v  = *(const float4*)(selfres + (size_t)w * 128 + c);
  float4 a1 = *(const float4*)(avec + c);
  float4 a2 = *(const float4*)(avec + 128 + c);
  float p = v.x * (a1.x + a2.x) + v.y * (a1.y + a2.y) +
            v.z * (a1.z + a2.z) + v.w * (a1.w + a2.w);
  p = wave_sum(p);
  if (lane == 0) logits[w] = p > 0.f ? p : 0.2f * p;
}

// logit = lrelu(selfres[idx[e]] . a[0:128] + edgev[e] . a[128:256])
__global__ __launch_bounds__(256) void att_edge_kernel(
    const float* __restrict__ selfres, const int* __restrict__ idx,
    const float* __restrict__ edgev, const float* __restrict__ avec,
    float* __restrict__ logits, int E)
{
  const int w = (blockIdx.x * blockDim.x + threadIdx.x) >> 5;
  const int lane = threadIdx.x & 31;
  if (w >= E) return;
  const int c = lane * 4;
  float4 v1 = *(const float4*)(selfres + (size_t)idx[w] * 128 + c);
  float4 v2 = *(const float4*)(edgev + (size_t)w * 128 + c);
  float4 a1 = *(const float4*)(avec + c);
  float4 a2 = *(const float4*)(avec + 128 + c);
  float p = v1.x * a1.x + v1.y * a1.y + v1.z * a1.z + v1.w * a1.w +
            v2.x * a2.x + v2.y * a2.y + v2.z * a2.z + v2.w * a2.w;
  p = wave_sum(p);
  if (lane == 0) logits[w] = p > 0.f ? p : 0.2f * p;
}

// ---------------------------------------------------------------------------
// Global softmax statistics (two-stage max, two-stage sum-exp)
// ---------------------------------------------------------------------------
__global__ __launch_bounds__(256) void reduce_max_partial_kernel(
    const float* __restrict__ v, int n, float* __restrict__ part)
{
  __shared__ float sm[8];
  float m = -3.402823e38f;
  for (int i = blockIdx.x * blockDim.x + threadIdx.x; i < n;
       i += gridDim.x * blockDim.x)
    m = fmaxf(m, v[i]);
#pragma unroll
  for (int o = 16; o > 0; o >>= 1) m = fmaxf(m, __shfl_xor(m, o, 32));
  const int lane = threadIdx.x & 31, wv = threadIdx.x >> 5;
  if (lane == 0) sm[wv] = m;
  __syncthreads();
  if (threadIdx.x < 32) {
    m = (lane < 8) ? sm[lane] : -3.402823e38f;
#pragma unroll
    for (int o = 4; o > 0; o >>= 1) m = fmaxf(m, __shfl_xor(m, o, 32));
    if (lane == 0) part[blockIdx.x] = m;
  }
}

__global__ __launch_bounds__(1024) void reduce_max_final_kernel(
    const float* __restrict__ part, float* __restrict__ scal)
{
  __shared__ float sm[32];
  float m = part[threadIdx.x];
#pragma unroll
  for (int o = 16; o > 0; o >>= 1) m = fmaxf(m, __shfl_xor(m, o, 32));
  const int lane = threadIdx.x & 31, wv = threadIdx.x >> 5;
  if (lane == 0) sm[wv] = m;
  __syncthreads();
  if (threadIdx.x < 32) {
    m = sm[lane];
#pragma unroll
    for (int o = 16; o > 0; o >>= 1) m = fmaxf(m, __shfl_xor(m, o, 32));
    if (lane == 0) scal[0] = m;
  }
}

__global__ __launch_bounds__(256) void reduce_sum_partial_kernel(
    const float* __restrict__ v, int n, const float* __restrict__ scal,
    float* __restrict__ part)
{
  __shared__ float sm[8];
  const float gmax = scal[0];
  float s = 0.f;
  for (int i = blockIdx.x * blockDim.x + threadIdx.x; i < n;
       i += gridDim.x * blockDim.x)
    s += expf(v[i] - gmax);
  s = wave_sum(s);
  const int lane = threadIdx.x & 31, wv = threadIdx.x >> 5;
  if (lane == 0) sm[wv] = s;
  __syncthreads();
  if (threadIdx.x < 32) {
    s = (lane < 8) ? sm[lane] : 0.f;
#pragma unroll
    for (int o = 4; o > 0; o >>= 1) s += __shfl_xor(s, o, 32);
    if (lane == 0) part[blockIdx.x] = s;
  }
}

__global__ __launch_bounds__(1024) void reduce_sum_final_kernel(
    const float* __restrict__ part, float* __restrict__ scal)
{
  __shared__ float sm[32];
  float s = part[threadIdx.x];
#pragma unroll
  for (int o = 16; o > 0; o >>= 1) s += __shfl_xor(s, o, 32);
  const int lane = threadIdx.x & 31, wv = threadIdx.x >> 5;
  if (lane == 0) sm[wv] = s;
  __syncthreads();
  if (threadIdx.x < 32) {
    s = sm[lane];
#pragma unroll
    for (int o = 16; o > 0; o >>= 1) s += __shfl_xor(s, o, 32);
    if (lane == 0) scal[1] = 1.0f / s;
  }
}

// ---------------------------------------------------------------------------
// Aggregation
// ---------------------------------------------------------------------------
__global__ __launch_bounds__(256) void init_out_kernel(
    const float* __restrict__ selfres, const float* __restrict__ logits,
    const float* __restrict__ scal, float* __restrict__ out)
{
  const int idx = blockIdx.x * 256 + threadIdx.x;  // R*128 total, exact grid
  const int r = idx >> 7;
  const float wt = expf(logits[r] - scal[0]) * scal[1];
  out[idx] = wt * selfres[idx];
}

// wave-per-edge coalesced scatter-add of softmax-weighted edge vectors
__global__ __launch_bounds__(256) void scatter_kernel(
    const float* __restrict__ V, const int* __restrict__ dst,
    const float* __restrict__ logits, const float* __restrict__ scal,
    float* __restrict__ out, int E)
{
  const int w = (blockIdx.x * blockDim.x + threadIdx.x) >> 5;
  const int lane = threadIdx.x & 31;
  if (w >= E) return;
  const float wt = expf(logits[w] - scal[0]) * scal[1];
  const float* vrow = V + (size_t)w * 128;
  float* orow = out + (size_t)dst[w] * 128;
#pragma unroll
  for (int i = 0; i < 4; ++i) {
    const int c = lane + 32 * i;
    atomicAdd(&orow[c], wt * vrow[c]);
  }
}

__global__ __launch_bounds__(256) void elu_kernel(float* __restrict__ out)
{
  const int idx = blockIdx.x * 256 + threadIdx.x;
  const float x = out[idx];
  out[idx] = x > 0.f ? x : (expf(x) - 1.0f);
}

// ---------------------------------------------------------------------------
extern "C" void kernel_launch(void* const* d_in, const int* in_sizes, int n_in,
                              void* d_out, int out_size, void* d_ws,
                              size_t ws_size, hipStream_t stream)
{
  (void)in_sizes; (void)n_in; (void)out_size; (void)ws_size;
  const float* resources  = (const float*)d_in[0];
  const float* operations = (const float*)d_in[1];
  const float* eattr      = (const float*)d_in[2];
  const int*   need_src   = (const int*)d_in[3];
  const int*   need_dst   = (const int*)d_in[4];
  const int*   same_src   = (const int*)d_in[5];
  const int*   same_dst   = (const int*)d_in[6];
  const float* W_self     = (const float*)d_in[7];
  const float* W_res      = (const float*)d_in[8];
  const float* W_op       = (const float*)d_in[9];
  const float* a_self     = (const float*)d_in[10];
  const float* a_op       = (const float*)d_in[11];
  const float* a_res      = (const float*)d_in[12];
  float* out = (float*)d_out;
  float* ws  = (float*)d_ws;

  // workspace layout (floats)
  float* self_res = ws;                               // R*128
  float* ops_edge = self_res + (size_t)R_N * 128;     // E1*128
  float* res1     = ops_edge + (size_t)E1_N * 128;    // E2*128
  float* logits   = res1 + (size_t)E2_N * 128;        // NT
  float* part     = logits + (size_t)NT_N;            // 2*NPART
  float* scal     = part + 2 * NPART;                 // [gmax, inv_sum]

  // 1-3: WMMA GEMMs (fp32, 16x16x4)
  gemm_self_kernel<<<R_N / 16, 256, 0, stream>>>(resources, W_self, self_res);
  gemm_op_kernel<<<E1_N / 16, 256, 0, stream>>>(operations, eattr, need_src,
                                                W_op, ops_edge);
  gemm_res_kernel<<<E2_N / 16, 256, 0, stream>>>(resources, same_src, W_res,
                                                 res1);

  // 4-6: attention logits
  att_self_kernel<<<R_N / 8, 256, 0, stream>>>(self_res, a_self, logits, R_N);
  att_edge_kernel<<<E1_N / 8, 256, 0, stream>>>(self_res, need_dst, ops_edge,
                                                a_op, logits + R_N, E1_N);
  att_edge_kernel<<<E2_N / 8, 256, 0, stream>>>(self_res, same_dst, res1,
                                                a_res, logits + R_N + E1_N,
                                                E2_N);

  // 7-10: global softmax statistics
  reduce_max_partial_kernel<<<NPART, 256, 0, stream>>>(logits, NT_N, part);
  reduce_max_final_kernel<<<1, 1024, 0, stream>>>(part, scal);
  reduce_sum_partial_kernel<<<NPART, 256, 0, stream>>>(logits, NT_N, scal,
                                                       part + NPART);
  reduce_sum_final_kernel<<<1, 1024, 0, stream>>>(part + NPART, scal);

  // 11-14: aggregate + activation
  init_out_kernel<<<(R_N * 128) / 256, 256, 0, stream>>>(self_res, logits,
                                                         scal, out);
  scatter_kernel<<<E1_N / 8, 256, 0, stream>>>(ops_edge, need_dst,
                                               logits + R_N, scal, out, E1_N);
  scatter_kernel<<<E2_N / 8, 256, 0, stream>>>(res1, same_dst,
                                               logits + R_N + E1_N, scal, out,
                                               E2_N);
  elu_kernel<<<(R_N * 128) / 256, 256, 0, stream>>>(out);
}